// Simulator_6803228197714
// MI455X (gfx1250) — compile-verified
//
#include <hip/hip_runtime.h>
#include <hip/hip_bf16.h>
#include <stdint.h>

#define N_AGENTS   4096
#define SROW       10          // normalized state row: pos(2) vel(2) acc(2) dest(2) tau pref
#define QCOLS      21
#define TILE       256         // j-agents per LDS tile
#define SPLIT      16          // threads cooperating per i-agent (divides 32 -> stays in one wave)
#define BLOCK      256
#define AGENTS_PER_BLOCK (BLOCK / SPLIT)   // 16
#define NTILES     (N_AGENTS / TILE)       // 16

__device__ __constant__ float kDT      = 0.04f;                 // 0.4 / 10
__device__ __constant__ float kTAU     = 0.5f;
__device__ __constant__ float kV0      = 2.1f;
__device__ __constant__ float kNEGL2S  = -4.8089834696298783f;  // -log2(e)/sigma, sigma=0.3
__device__ __constant__ float kPOTDT   = 0.4f;
__device__ __constant__ float kCOSPHI  = -0.17364817766693041f; // cos(100 deg)
__device__ __constant__ float kOOV     = 0.5f;
__device__ __constant__ float kFD      = 0.001f;
__device__ __constant__ float kINVFD   = 1000.0f;
__device__ __constant__ float kMAXMULT = 1.3f;
__device__ __constant__ float kEPS     = 1e-8f;

__device__ __forceinline__ float fsqrt(float x) { return __builtin_amdgcn_sqrtf(x); }
__device__ __forceinline__ float fexp2(float x) { return __builtin_amdgcn_exp2f(x); }

// ---------------------------------------------------------------------------
// CDNA5 async global -> LDS copy (tracked by ASYNCcnt, not LOADcnt).
// lds addr = low 32 bits of the flat address of a __shared__ object (ISA 10.2:
// LDS aperture lives entirely in addr[63:32]; addr[31:0] is the LDS offset).
// GVS mode: mem_addr = SADDR(sgpr pair) + VADDR(32-bit vgpr byte offset).
// ---------------------------------------------------------------------------
__device__ __forceinline__ uint32_t lds_addr_of(const void* p) {
  return (uint32_t)(uintptr_t)p;
}

__device__ __forceinline__ void async_load_b128(void* lds_dst, const void* gbase,
                                                int byte_off) {
  asm volatile("global_load_async_to_lds_b128 %0, %1, %2"
               :: "v"(lds_addr_of(lds_dst)), "v"(byte_off), "s"(gbase)
               : "memory");
}
__device__ __forceinline__ void async_load_b32(void* lds_dst, const void* gbase,
                                               int byte_off) {
  asm volatile("global_load_async_to_lds_b32 %0, %1, %2"
               :: "v"(lds_addr_of(lds_dst)), "v"(byte_off), "s"(gbase)
               : "memory");
}
__device__ __forceinline__ void wait_async0() {
  asm volatile("s_wait_asynccnt 0" ::: "memory");
}

// social-force pair potential  V0 * exp(-b / sigma) = V0 * exp2(b * kNEGL2S),
// with the |rab| norm supplied by the caller (shared with the FOV test).
__device__ __forceinline__ float pot_n1(float n1, float rx, float ry,
                                        float ox, float oy, float c) {
  float qx = rx - ox, qy = ry - oy;
  float n2 = fsqrt(qx * qx + qy * qy);
  float s  = (n1 + n2) * (n1 + n2) - c * c;
  s = fmaxf(s, 0.0f);
  float b = 0.5f * fsqrt(s);
  return kV0 * fexp2(b * kNEGL2S);
}
__device__ __forceinline__ float pot(float rx, float ry, float ox, float oy, float c) {
  return pot_n1(fsqrt(rx * rx + ry * ry), rx, ry, ox, oy, c);
}

// ---------------------------------------------------------------------------
// init: normalize raw (N,6) state into (N,10), copy dest + queues into ws
// ---------------------------------------------------------------------------
__global__ __launch_bounds__(256) void sfm_init(
    const float* __restrict__ s6, const float* __restrict__ dest_in,
    const float* __restrict__ q_in, float* __restrict__ state,
    float* __restrict__ dest, float* __restrict__ queue) {
  int i = blockIdx.x * blockDim.x + threadIdx.x;
  if (i >= N_AGENTS) return;
  const float* S = s6 + i * 6;
  float vx = S[2], vy = S[3];
  float* O = state + i * SROW;
  O[0] = S[0]; O[1] = S[1]; O[2] = vx; O[3] = vy;
  O[4] = 0.0f; O[5] = 0.0f; O[6] = S[4]; O[7] = S[5];
  O[8] = kTAU; O[9] = fsqrt(vx * vx + vy * vy);
#pragma unroll
  for (int k = 0; k < 4; ++k) dest[i * 4 + k] = dest_in[i * 4 + k];
#pragma unroll
  for (int k = 0; k < QCOLS; ++k) queue[i * QCOLS + k] = q_in[i * QCOLS + k];
}

// ---------------------------------------------------------------------------
// aux pack: per-agent interaction record  {x, y, c*ex, c*ey} (+ c), c=POT_DT*spd
// ---------------------------------------------------------------------------
__global__ __launch_bounds__(256) void sfm_aux(
    const float* __restrict__ state, float4* __restrict__ aux4,
    float* __restrict__ auxc) {
  int j = blockIdx.x * blockDim.x + threadIdx.x;
  if (j >= N_AGENTS) return;
  const float* S = state + j * SROW;
  float px = S[0], py = S[1], vx = S[2], vy = S[3], dx = S[6], dy = S[7];
  float spd = fsqrt(vx * vx + vy * vy);
  float c   = kPOTDT * spd;
  float ddx = dx - px, ddy = dy - py;
  float dn  = fmaxf(fsqrt(ddx * ddx + ddy * ddy), kEPS);
  aux4[j] = make_float4(px, py, c * ddx / dn, c * ddy / dn);
  auxc[j] = c;
}

// ---------------------------------------------------------------------------
// fused step: O(N^2) forces (async-LDS double-buffered j-tiles) + integrate
// + destination-queue logic.  SPLIT threads per agent, reduced with wave32
// xor-shuffles; lane t==0 commits the per-agent update.
// ---------------------------------------------------------------------------
__global__ __launch_bounds__(BLOCK) void sfm_step(
    const float* __restrict__ stateOld, const float4* __restrict__ aux4,
    const float* __restrict__ auxc, float* __restrict__ dest,
    float* __restrict__ queue, float* __restrict__ stateNew) {
  __shared__ float4 t4[2][TILE];
  __shared__ float  tc[2][TILE];

  const int tid = threadIdx.x;
  const int t   = tid & (SPLIT - 1);
  const int i   = blockIdx.x * AGENTS_PER_BLOCK + (tid / SPLIT);

  // own agent data needed by all SPLIT cooperating lanes
  const float* Si = stateOld + i * SROW;
  float px = Si[0], py = Si[1];
  float ddx = Si[6] - px, ddy = Si[7] - py;
  float dn  = fmaxf(fsqrt(ddx * ddx + ddy * ddy), kEPS);
  float exi = ddx / dn, eyi = ddy / dn;

  float Fx = 0.0f, Fy = 0.0f;

  // prologue: tile 0 -> buffer 0 (one 16B record + one 4B scalar per lane)
  async_load_b128(&t4[0][tid], aux4, tid * 16);
  async_load_b32(&tc[0][tid], auxc, tid * 4);

  for (int tt = 0; tt < NTILES; ++tt) {
    const int cur = tt & 1;
    wait_async0();          // my lane's loads for tile tt are in LDS
    __syncthreads();        // everyone's loads done; everyone done reading buf 1-cur
    if (tt + 1 < NTILES) {  // prefetch next tile into the other buffer
      int jn = (tt + 1) * TILE + tid;
      async_load_b128(&t4[1 - cur][tid], aux4, jn * 16);
      async_load_b32(&tc[1 - cur][tid], auxc, jn * 4);
    }
    const int jbase = tt * TILE;
#pragma unroll 4
    for (int k = t; k < TILE; k += SPLIT) {
      const int j = jbase + k;
      float4 a = t4[cur][k];
      float  c = tc[cur][k];
      float rx = px - a.x, ry = py - a.y;
      float n1 = fsqrt(rx * rx + ry * ry);
      float v  = pot_n1(n1, rx, ry, a.z, a.w, c);
      float vx = pot(rx + kFD, ry, a.z, a.w, c);
      float vy = pot(rx, ry + kFD, a.z, a.w, c);
      // fov: f = -rab ; in_sight = e . f > |f| * cosphi ; diagonal -> w = 0
      float w  = ((exi * (-rx) + eyi * (-ry)) > n1 * kCOSPHI) ? 1.0f : kOOV;
      w = (j == i) ? 0.0f : w;
      Fx += w * (v - vx) * kINVFD;   // -(dV/dx)
      Fy += w * (v - vy) * kINVFD;
    }
  }

  // reduce the SPLIT partial sums (contiguous lanes inside one wave32)
  Fx += __shfl_xor(Fx, 1, 32); Fy += __shfl_xor(Fy, 1, 32);
  Fx += __shfl_xor(Fx, 2, 32); Fy += __shfl_xor(Fy, 2, 32);
  Fx += __shfl_xor(Fx, 4, 32); Fy += __shfl_xor(Fy, 4, 32);
  Fx += __shfl_xor(Fx, 8, 32); Fy += __shfl_xor(Fy, 8, 32);

  if (t == 0) {
    float vx = Si[2], vy = Si[3], ax = Si[4], ay = Si[5];
    float tau = Si[8], pref = Si[9];
    // F = F0 + sum_j w * f_ab
    float Fxt = (pref * exi - vx) / tau + Fx;
    float Fyt = (pref * eyi - vy) / tau + Fy;
    float npx = px + vx * kDT + 0.5f * Fxt * kDT * kDT;
    float npy = py + vy * kDT + 0.5f * Fyt * kDT * kDT;
    float nvx = vx + 0.5f * (ax + Fxt) * kDT;
    float nvy = vy + 0.5f * (ay + Fyt) * kDT;

    // distance from NEW pos to current destination segment
    float* D = dest + i * 4;
    float p1x = D[0], p1y = D[1], p2x = D[2], p2y = D[3];
    float sx = p2x - p1x, sy = p2y - p1y;
    float tseg = ((npx - p1x) * sx + (npy - p1y) * sy) /
                 fmaxf(sx * sx + sy * sy, kEPS);
    tseg = fminf(fmaxf(tseg, 0.0f), 1.0f);
    float cxp = p1x + tseg * sx - npx, cyp = p1y + tseg * sy - npy;
    bool reached = (cxp * cxp + cyp * cyp) < (0.05f * 0.05f);

    float* Q = queue + i * QCOLS;
    if (reached) {           // in-place left shift by one 4-col entry
      Q[0] += 1.0f;
#pragma unroll
      for (int k = 1; k <= 16; ++k) Q[k] = Q[k + 4];
      // Q[17..20] stay (last entry repeats)
    }
    float q1 = Q[1], q2 = Q[2], q3 = Q[3], q4 = Q[4];
    D[0] = q1; D[1] = q2; D[2] = q3; D[3] = q4;

    // nearest point of NEW destination segment to OLD position
    float gx = q3 - q1, gy = q4 - q2;
    float t2 = ((px - q1) * gx + (py - q2) * gy) /
               fmaxf(gx * gx + gy * gy, kEPS);
    t2 = fminf(fmaxf(t2, 0.0f), 1.0f);
    float npdx = q1 + t2 * gx, npdy = q2 + t2 * gy;

    // speed clamp
    float ds = fsqrt(nvx * nvx + nvy * nvy);
    float factor = fminf(pref * kMAXMULT / fmaxf(ds, kEPS), 1.0f);

    float* O = stateNew + i * SROW;
    O[0] = npx; O[1] = npy;
    O[2] = nvx * factor; O[3] = nvy * factor;
    O[4] = Fxt; O[5] = Fyt;
    O[6] = npdx; O[7] = npdy;
    O[8] = tau; O[9] = pref;
  }
}

extern "C" void kernel_launch(void* const* d_in, const int* in_sizes, int n_in,
                              void* d_out, int out_size, void* d_ws, size_t ws_size,
                              hipStream_t stream) {
  (void)in_sizes; (void)n_in; (void)out_size; (void)ws_size;
  const float* in_state = (const float*)d_in[0];   // (N,6)
  const float* in_dest  = (const float*)d_in[1];   // (N,4)
  const float* in_queue = (const float*)d_in[2];   // (N,21)

  float* ws = (float*)d_ws;
  float*  stateA = ws;                       // N*10
  float*  stateB = ws + N_AGENTS * 10;       // N*10
  float*  dest   = ws + N_AGENTS * 20;       // N*4
  float*  queue  = ws + N_AGENTS * 24;       // N*21
  float4* aux4   = (float4*)(ws + N_AGENTS * 45);  // N*4 (16B aligned)
  float*  auxc   = ws + N_AGENTS * 49;       // N

  sfm_init<<<N_AGENTS / 256, 256, 0, stream>>>(in_state, in_dest, in_queue,
                                               stateA, dest, queue);

  float* cur = stateA;
  float* nxt = stateB;
  for (int s = 0; s < 10; ++s) {
    sfm_aux<<<N_AGENTS / 256, 256, 0, stream>>>(cur, aux4, auxc);
    float* out = (s == 9) ? (float*)d_out : nxt;
    sfm_step<<<(N_AGENTS * SPLIT) / BLOCK, BLOCK, 0, stream>>>(
        cur, aux4, auxc, dest, queue, out);
    float* tmp = cur; cur = nxt; nxt = tmp;
  }
}